// GraphConv_1443109011527
// MI455X (gfx1250) — compile-verified
//
#include <hip/hip_runtime.h>

typedef __attribute__((ext_vector_type(2))) float v2f;
typedef __attribute__((ext_vector_type(8))) float v8f;

#define DFEAT 64

// ---------------- zero workspace ----------------
__global__ void zero_f32_kernel(float* __restrict__ p, int n) {
    int i = blockIdx.x * blockDim.x + threadIdx.x;
    if (i < n) p[i] = 0.0f;
}

// ---------------- in-degree via float atomics ----------------
__global__ void degree_kernel(const int* __restrict__ dst, float* __restrict__ deg, int E) {
    int e = blockIdx.x * blockDim.x + threadIdx.x;
    if (e < E) atomicAdd(&deg[dst[e]], 1.0f);
}

// ---------------- norm = clip(deg,1)^-0.5 (in place) ----------------
__global__ void norm_kernel(float* __restrict__ deg, int n) {
    int i = blockIdx.x * blockDim.x + threadIdx.x;
    if (i < n) deg[i] = rsqrtf(fmaxf(deg[i], 1.0f));
}

// ---------------- scatter-add: agg[dst] += feat[src] * norm[src] ----------------
// One wave (32 lanes) per edge; lane handles features lane and lane+32.
__global__ void scatter_kernel(const float* __restrict__ feat,
                               const int* __restrict__ src,
                               const int* __restrict__ dst,
                               const float* __restrict__ norm,
                               float* __restrict__ agg, int E) {
    int gtid = blockIdx.x * blockDim.x + threadIdx.x;
    int e    = gtid >> 5;
    int lane = gtid & 31;
    if (e >= E) return;
    int s = src[e];
    int d = dst[e];
    float ns = norm[s];
    float v0 = feat[(size_t)s * DFEAT + lane]      * ns;
    float v1 = feat[(size_t)s * DFEAT + lane + 32] * ns;
    atomicAdd(&agg[(size_t)d * DFEAT + lane],      v0);
    atomicAdd(&agg[(size_t)d * DFEAT + lane + 32], v1);
}

// ---------------- out = (agg @ W) * norm[:,None] + bias, via f32 WMMA ----------------
// One wave computes one 16x16 output tile. K-loop: 16 x V_WMMA_F32_16X16X4_F32.
__global__ void gcn_matmul_wmma_kernel(const float* __restrict__ agg,
                                       const float* __restrict__ W,     // 64x64 row-major
                                       const float* __restrict__ norm,
                                       const float* __restrict__ bias,
                                       float* __restrict__ out,
                                       int nRowTiles) {
    __shared__ float Ws[DFEAT * DFEAT];  // 16 KB of the 320 KB WGP LDS
    for (int i = threadIdx.x; i < DFEAT * DFEAT; i += blockDim.x) Ws[i] = W[i];
    __syncthreads();

    const int lane   = threadIdx.x & 31;
    const int waveIb = threadIdx.x >> 5;
    const int waveId = blockIdx.x * (blockDim.x >> 5) + waveIb;
    const int totalWaves = nRowTiles * (DFEAT / 16);
    if (waveId >= totalWaves) return;   // wave-uniform: EXEC stays all-ones for WMMA

    const int rowTile = waveId >> 2;        // DFEAT/16 == 4 col tiles
    const int colTile = waveId & 3;
    const int rowBase = rowTile * 16;
    const int colBase = colTile * 16;

    const int half = lane >> 4;   // 0: lanes 0-15, 1: lanes 16-31
    const int l16  = lane & 15;

    // A (16x4 f32): lane holds M=l16, K = half*2 + {0,1}
    const float* Arow = agg + (size_t)(rowBase + l16) * DFEAT;

    v8f c = {};
#pragma unroll
    for (int kk = 0; kk < DFEAT; kk += 4) {
        const int ka = kk + half * 2;
        v2f a, b;
        a.x = Arow[ka];
        a.y = Arow[ka + 1];
        // B (4x16 f32): lane holds N=l16, K = half*2 + {vgpr}
        b.x = Ws[(ka    ) * DFEAT + colBase + l16];
        b.y = Ws[(ka + 1) * DFEAT + colBase + l16];
        // 8-arg form: (neg_a, A, neg_b, B, c_mod, C, reuse_a, reuse_b)
        c = __builtin_amdgcn_wmma_f32_16x16x4_f32(false, a, false, b, (short)0, c,
                                                  false, false);
    }

    // D (16x16 f32): VGPR v -> row = v + half*8, col = l16
#pragma unroll
    for (int v = 0; v < 8; ++v) {
        const int row = rowBase + v + half * 8;
        const int col = colBase + l16;
        out[(size_t)row * DFEAT + col] = c[v] * norm[row] + bias[col];
    }
}

// ---------------- launcher ----------------
extern "C" void kernel_launch(void* const* d_in, const int* in_sizes, int n_in,
                              void* d_out, int out_size, void* d_ws, size_t ws_size,
                              hipStream_t stream) {
    const float* feat   = (const float*)d_in[0];
    const float* weight = (const float*)d_in[1];
    const float* bias   = (const float*)d_in[2];
    const int*   src    = (const int*)d_in[3];
    const int*   dst    = (const int*)d_in[4];
    float*       out    = (float*)d_out;

    const int N = in_sizes[0] / DFEAT;   // 50000
    const int E = in_sizes[3];           // 800000

    float* agg  = (float*)d_ws;                   // N*DFEAT floats
    float* norm = agg + (size_t)N * DFEAT;        // N floats (deg -> norm in place)

    // 1) zero agg + deg (contiguous)
    const int zn = N * DFEAT + N;
    zero_f32_kernel<<<(zn + 255) / 256, 256, 0, stream>>>(agg, zn);

    // 2) in-degree
    degree_kernel<<<(E + 255) / 256, 256, 0, stream>>>(dst, norm, E);

    // 3) norm = rsqrt(max(deg,1))
    norm_kernel<<<(N + 255) / 256, 256, 0, stream>>>(norm, N);

    // 4) scatter-add (one wave per edge)
    const long long sthreads = (long long)E * 32;
    scatter_kernel<<<(int)((sthreads + 255) / 256), 256, 0, stream>>>(feat, src, dst,
                                                                      norm, agg, E);

    // 5) WMMA GEMM + epilogue (N divisible by 16 for this problem)
    const int nRowTiles  = N / 16;                 // 3125
    const int totalWaves = nRowTiles * (DFEAT / 16);
    const int blocks     = (totalWaves + 7) / 8;   // 8 waves per 256-thread block
    gcn_matmul_wmma_kernel<<<blocks, 256, 0, stream>>>(agg, weight, norm, bias, out,
                                                       nRowTiles);
}